// DownscalingLoss_81423989997802
// MI455X (gfx1250) — compile-verified
//
#include <hip/hip_runtime.h>
#include <float.h>

// Problem geometry (fixed by reference setup_inputs)
#define IMG_H 512
#define IMG_W 512
#define WIN   7
#define OUT_H (IMG_H - WIN + 1)   // 506
#define OUT_W (IMG_W - WIN + 1)   // 506

typedef __attribute__((ext_vector_type(2))) float v2f;
typedef __attribute__((ext_vector_type(8))) float v8f;

// ---------------------------------------------------------------------------
// fp32 WMMA 16x16x4:  D(16x16) = A(16x4) * B(4x16) + C
// A layout (ISA 7.12.2): lanes 0-15 rows M=0..15; VGPR0 -> K = 2*(lane/16),
// VGPR1 -> K = 2*(lane/16)+1.  B: rows striped across lanes per VGPR, same
// half-wave split for K.  C/D: VGPR j -> rows j (lanes 0-15) / j+8 (16-31).
// ---------------------------------------------------------------------------
__device__ __forceinline__ v8f wmma4(v2f a, v2f b, v8f c) {
  return __builtin_amdgcn_wmma_f32_16x16x4_f32(
      /*neg_a=*/false, a, /*neg_b=*/false, b,
      /*c_mod=*/(short)0, c, /*reuse_a=*/false, /*reuse_b=*/false);
}

// banded ones: 1.0 if c <= m <= c+6 (7-wide box band)
__device__ __forceinline__ float bandf(int m, int c) {
  return (m >= c && m <= c + 6) ? 1.0f : 0.0f;
}

__device__ __forceinline__ float mapval(int map, float p, float t) {
  switch (map) {
    case 0: return p;
    case 1: return t;
    case 2: return p * p;
    case 3: return t * t;
    default: return p * t;
  }
}

__device__ __forceinline__ void atomicMaxF(float* addr, float v) {
  int* ia = (int*)addr;
  int old = __float_as_int(*addr);
  while (__int_as_float(old) < v) {
    int prev = atomicCAS(ia, old, __float_as_int(v));
    if (prev == old) break;
    old = prev;
  }
}

__device__ __forceinline__ void atomicMinF(float* addr, float v) {
  int* ia = (int*)addr;
  int old = __float_as_int(*addr);
  while (__int_as_float(old) > v) {
    int prev = atomicCAS(ia, old, __float_as_int(v));
    if (prev == old) break;
    old = prev;
  }
}

// ---------------------------------------------------------------------------
// Workspace layout:  sums[0]=sum|d|, sums[1]=sum d^2, sums[2]=sum SSIM (f64)
//                    then maxv[nimg], minv[nimg] (f32)
// ---------------------------------------------------------------------------
__global__ void init_ws(double* sums, float* maxv, float* minv, int nimg) {
  int i = threadIdx.x;
  if (i < 3) sums[i] = 0.0;
  if (i < nimg) { maxv[i] = -FLT_MAX; minv[i] = FLT_MAX; }
}

// ---------------------------------------------------------------------------
// Streaming stats: mae/mse partial sums + per-image max/min of yt.
// One block owns a contiguous chunk of a single image (clean max/min scope).
// float4 loads -> GLOBAL_LOAD_B128, HBM-bandwidth limited.
// ---------------------------------------------------------------------------
__global__ void __launch_bounds__(256)
stat_kernel(const float* __restrict__ ypIn, const float* __restrict__ ytIn,
            const float* __restrict__ mkIn, double* sums, float* maxv, float* minv,
            int chunksPerImg, int chunkElems) {
  const int HW = IMG_H * IMG_W;
  const int img   = blockIdx.x / chunksPerImg;
  const int chunk = blockIdx.x % chunksPerImg;
  const int base4 = (img * HW + chunk * chunkElems) >> 2;   // 16B aligned
  const int n4    = chunkElems >> 2;

  const float4* yp4 = (const float4*)ypIn;
  const float4* yt4 = (const float4*)ytIn;
  const float4* mk4 = (const float4*)mkIn;

  float sa = 0.f, sq = 0.f, mx = -FLT_MAX, mn = FLT_MAX;
  for (int i = threadIdx.x; i < n4; i += blockDim.x) {
    float4 p = yp4[base4 + i];
    float4 t = yt4[base4 + i];
    float4 m = mk4[base4 + i];
    float pp[4] = {p.x * m.x, p.y * m.y, p.z * m.z, p.w * m.w};
    float tt[4] = {t.x * m.x, t.y * m.y, t.z * m.z, t.w * m.w};
#pragma unroll
    for (int k = 0; k < 4; ++k) {
      float d = pp[k] - tt[k];
      sa += fabsf(d);
      sq += d * d;
      mx = fmaxf(mx, tt[k]);
      mn = fminf(mn, tt[k]);
    }
  }

  __shared__ float rA[256], rQ[256], rM[256], rN[256];
  int tid = threadIdx.x;
  rA[tid] = sa; rQ[tid] = sq; rM[tid] = mx; rN[tid] = mn;
  __syncthreads();
  for (int s = 128; s > 0; s >>= 1) {
    if (tid < s) {
      rA[tid] += rA[tid + s];
      rQ[tid] += rQ[tid + s];
      rM[tid] = fmaxf(rM[tid], rM[tid + s]);
      rN[tid] = fminf(rN[tid], rN[tid + s]);
    }
    __syncthreads();
  }
  if (tid == 0) {
    atomicAdd(&sums[0], (double)rA[0]);
    atomicAdd(&sums[1], (double)rQ[0]);
    atomicMaxF(&maxv[img], rM[0]);
    atomicMinF(&minv[img], rN[0]);
  }
}

// ---------------------------------------------------------------------------
// SSIM kernel.  One block = one 64x64 output macro-tile of one image.
// LDS: 70x70 haloed yp/yt tiles (stride 72 to dodge bank conflicts).
// Each wave computes two 16x16 output tiles:
//   stage 1 (horizontal 7-sum): H(22x16) = X(22x24) * Bband(24x16), done as
//     two 16-row WMMA groups x 6 K-chunks of V_WMMA_F32_16X16X4_F32
//   stage 2 (vertical 7-sum):   D(16x16) = Aband(16x24) * H(24x16), 6 chunks
// Band matrices zero the K=22..23 padding, so clamped halo reads are inert.
// ---------------------------------------------------------------------------
#define TSTR 72
__global__ void __launch_bounds__(256)
ssim_kernel(const float* __restrict__ ypIn, const float* __restrict__ ytIn,
            const float* __restrict__ mkIn, double* sums,
            const float* __restrict__ maxv, const float* __restrict__ minv) {
  __shared__ float tp[70 * TSTR];
  __shared__ float tt[70 * TSTR];
  __shared__ float scrAll[8 * 32 * 16];   // per-wave H scratch (32x16 each)
  __shared__ float red[256];

  const int HW  = IMG_H * IMG_W;
  const int img = blockIdx.x >> 6;
  const int mt  = blockIdx.x & 63;
  const int mr  = (mt >> 3) * 64;   // macro-tile origin (output coords)
  const int mc  = (mt & 7) * 64;

  // Cooperative haloed tile load: yp = y_pred*mask, yt = y_true*mask
  for (int i = threadIdx.x; i < 70 * 70; i += 256) {
    int r = i / 70, c = i % 70;
    int gr = mr + r; gr = gr > IMG_H - 1 ? IMG_H - 1 : gr;
    int gc = mc + c; gc = gc > IMG_W - 1 ? IMG_W - 1 : gc;
    int idx = img * HW + gr * IMG_W + gc;
    float m = mkIn[idx];
    tp[r * TSTR + c] = ypIn[idx] * m;
    tt[r * TSTR + c] = ytIn[idx] * m;
  }
  __syncthreads();

  const float drange = maxv[img] - minv[img];
  float C1 = 0.01f * drange; C1 *= C1;
  float C2 = 0.03f * drange; C2 *= C2;

  const int lane  = threadIdx.x & 31;
  const int wave  = threadIdx.x >> 5;
  const int lrow  = lane & 15;   // M (or N) index
  const int lhalf = lane >> 4;   // K half-wave split
  float* scr = &scrAll[wave * 32 * 16];

  const float inv49 = 1.0f / 49.0f;
  const float covn  = 49.0f / 48.0f;
  float localS = 0.f;

  for (int sti = 0; sti < 2; ++sti) {
    const int st = wave + sti * 8;        // subtile 0..15 in 4x4 grid
    const int r0 = (st >> 2) * 16;
    const int c0 = (st & 3) * 16;
    v8f res[5];

#pragma unroll
    for (int map = 0; map < 5; ++map) {
      // ---- stage 1: horizontal banded matmul, two 16-row groups ----
#pragma unroll
      for (int g = 0; g < 2; ++g) {
        v8f acc = {};
        int row = r0 + g * 16 + lrow;
        if (row > 69) row = 69;           // clamp; killed by stage-2 band
        const float* tpr = &tp[row * TSTR];
        const float* ttr = &tt[row * TSTR];
#pragma unroll
        for (int kk = 0; kk < 6; ++kk) {
          int k0 = 4 * kk + 2 * lhalf;
          int ca = c0 + k0, cb = ca + 1;
          if (ca > 69) ca = 69;           // clamp; band coeff is 0 there
          if (cb > 69) cb = 69;
          v2f a, b;
          a.x = mapval(map, tpr[ca], ttr[ca]);
          a.y = mapval(map, tpr[cb], ttr[cb]);
          b.x = bandf(k0, lrow);
          b.y = bandf(k0 + 1, lrow);
          acc = wmma4(a, b, acc);
        }
#pragma unroll
        for (int j = 0; j < 8; ++j)
          scr[(g * 16 + j + 8 * lhalf) * 16 + lrow] = acc[j];
      }
      // ---- stage 2: vertical banded matmul ----
      v8f acc2 = {};
#pragma unroll
      for (int kk = 0; kk < 6; ++kk) {
        int j0 = 4 * kk + 2 * lhalf;
        v2f a, b;
        a.x = bandf(j0, lrow);
        a.y = bandf(j0 + 1, lrow);
        b.x = scr[j0 * 16 + lrow];
        b.y = scr[(j0 + 1) * 16 + lrow];
        acc2 = wmma4(a, b, acc2);
      }
      res[map] = acc2;
    }

    // ---- SSIM rational expression on the 8 (row,col) elements per lane ----
#pragma unroll
    for (int j = 0; j < 8; ++j) {
      int R  = mr + r0 + j + 8 * lhalf;
      int Cc = mc + c0 + lrow;
      if (R < OUT_H && Cc < OUT_W) {
        float ux  = res[0][j] * inv49;
        float uy  = res[1][j] * inv49;
        float vx  = covn * (res[2][j] * inv49 - ux * ux);
        float vy  = covn * (res[3][j] * inv49 - uy * uy);
        float vxy = covn * (res[4][j] * inv49 - ux * uy);
        float Sv = ((2.f * ux * uy + C1) * (2.f * vxy + C2)) /
                   ((ux * ux + uy * uy + C1) * (vx + vy + C2));
        localS += Sv;
      }
    }
  }

  red[threadIdx.x] = localS;
  __syncthreads();
  for (int s = 128; s > 0; s >>= 1) {
    if (threadIdx.x < s) red[threadIdx.x] += red[threadIdx.x + s];
    __syncthreads();
  }
  if (threadIdx.x == 0) atomicAdd(&sums[2], (double)red[0]);
}

// ---------------------------------------------------------------------------
__global__ void finalize_kernel(const double* __restrict__ sums, float* out,
                                int ntotal, int nimg) {
  if (threadIdx.x == 0 && blockIdx.x == 0) {
    double N = (double)ntotal;
    double mae = sums[0] / N;
    double mse = sums[1] / N;
    double smean = sums[2] / ((double)nimg * (double)(OUT_H * OUT_W));
    double sloss = 1.0 - smean;
    double total = mae + 0.5 * mse + 0.2 * sloss;
    out[0] = (float)total;
    out[1] = (float)mae;
    out[2] = (float)mse;
    out[3] = (float)sloss;
  }
}

// ---------------------------------------------------------------------------
extern "C" void kernel_launch(void* const* d_in, const int* in_sizes, int n_in,
                              void* d_out, int out_size, void* d_ws, size_t ws_size,
                              hipStream_t stream) {
  const float* yp = (const float*)d_in[0];
  const float* yt = (const float*)d_in[1];
  const float* mk = (const float*)d_in[2];
  const int n    = in_sizes[0];
  const int nimg = n / (IMG_H * IMG_W);   // 48

  double* sums = (double*)d_ws;
  float*  maxv = (float*)((char*)d_ws + 3 * sizeof(double));
  float*  minv = maxv + nimg;

  init_ws<<<1, 64, 0, stream>>>(sums, maxv, minv, nimg);

  const int chunksPerImg = 64;
  const int chunkElems   = (IMG_H * IMG_W) / chunksPerImg;  // 4096
  stat_kernel<<<nimg * chunksPerImg, 256, 0, stream>>>(
      yp, yt, mk, sums, maxv, minv, chunksPerImg, chunkElems);

  // 64 macro-tiles (8x8 of 64x64 outputs) per image
  ssim_kernel<<<nimg * 64, 256, 0, stream>>>(yp, yt, mk, sums, maxv, minv);

  finalize_kernel<<<1, 32, 0, stream>>>(sums, (float*)d_out, n, nimg);
}